// DWT_24017457119247
// MI455X (gfx1250) — compile-verified
//
#include <hip/hip_runtime.h>

// ---------------------------------------------------------------------------
// 3D Haar DWT, single level, f32.  x:[2,32,32,128,128] -> LLL:[2,32,16,64,64]
// ++ H_all:[2,224,16,64,64] (concat LLH,LHL,LHH,HLL,HLH,HHL,HHH on channels).
// Memory-bound (256 MiB traffic, ~11.5 us at 23.3 TB/s). No WMMA on purpose:
// AI ~0.6 flop/byte. CDNA5 path used: async global->LDS bulk loads + asynccnt.
// ---------------------------------------------------------------------------

#ifndef __has_builtin
#define __has_builtin(x) 0
#endif

#if __has_builtin(__builtin_amdgcn_global_load_async_to_lds_b128)
#define HAVE_ASYNC_LDS 1
#else
#define HAVE_ASYNC_LDS 0
#endif

#if __has_builtin(__builtin_amdgcn_s_wait_asynccnt)
#define HAVE_WAIT_ASYNCCNT 1
#else
#define HAVE_WAIT_ASYNCCNT 0
#endif

// Builtin signature (from clang diagnostic):
//   void __builtin_amdgcn_global_load_async_to_lds_b128(
//       v4i addrspace(1)* src, v4i addrspace(3)* dst, imm int offset, imm int cpol)
typedef int v4i_ __attribute__((vector_size(4 * sizeof(int))));
typedef __attribute__((address_space(1))) v4i_* as1_v4i;
typedef __attribute__((address_space(3))) v4i_* as3_v4i;

#define BC_   64    // B*C fused
#define DIN_  32
#define HIN_  128
#define WIN_  128
#define DOUT_ 16
#define HOUT_ 64
#define WOUT_ 64

__global__ __launch_bounds__(256) void haar3d_kernel(const float* __restrict__ x,
                                                     float* __restrict__ out) {
    __shared__ __align__(16) float tile[2 * 16 * WIN_];   // 16 KB: 2 D-planes x 16 H rows x 128 W

    const int id    = blockIdx.x;          // [0, 64*16*8)
    const int tileY = id & 7;              // 8-output-row tile index along H
    const int dz    = (id >> 3) & 15;      // output D index
    const int bc    = id >> 7;             // fused batch*channel

    // ---- stage input tile: planes d=2dz..2dz+1, rows h=16*tileY..+15, all W ----
    const size_t srcBase =
        ((size_t)(bc * DIN_ + 2 * dz) * HIN_ + (size_t)(16 * tileY)) * WIN_;

#pragma unroll
    for (int i = 0; i < 4; ++i) {
        const int q  = threadIdx.x + 256 * i;   // 0..1023 chunks of 4 floats
        const int w4 = (q & 31) * 4;            // W offset
        const int h  = (q >> 5) & 15;           // row within staged slab
        const int d  = q >> 9;                  // which D plane
        const float* g = x + srcBase + (size_t)(d * HIN_ + h) * WIN_ + w4;
        float* l = &tile[(d * 16 + h) * WIN_ + w4];
#if HAVE_ASYNC_LDS
        __builtin_amdgcn_global_load_async_to_lds_b128(
            (as1_v4i)(void*)g, (as3_v4i)(void*)l, /*offset=*/0, /*cpol=*/0);
#else
        *(float4*)l = *(const float4*)g;
#endif
    }
#if HAVE_ASYNC_LDS
#if HAVE_WAIT_ASYNCCNT
    __builtin_amdgcn_s_wait_asynccnt(0);
#else
    asm volatile("s_wait_asynccnt 0" ::: "memory");
#endif
#endif
    __syncthreads();

    // ---- compute: thread -> (output row in tile, pair of output cols) ----
    const int t   = threadIdx.x;
    const int p   = t & 31;                 // output dx pair: dx = 2p, 2p+1
    const int row = t >> 5;                 // 0..7
    const int dyG = tileY * 8 + row;        // global output H index

    const float4 a00 = *(const float4*)&tile[( 0 + 2 * row    ) * WIN_ + 4 * p];
    const float4 a01 = *(const float4*)&tile[( 0 + 2 * row + 1) * WIN_ + 4 * p];
    const float4 a10 = *(const float4*)&tile[(16 + 2 * row    ) * WIN_ + 4 * p];
    const float4 a11 = *(const float4*)&tile[(16 + 2 * row + 1) * WIN_ + 4 * p];

    // W-axis butterfly (two output columns: x-lane from .x/.y, y-lane from .z/.w)
    const float s00x = a00.x + a00.y, s00y = a00.z + a00.w;
    const float q00x = a00.x - a00.y, q00y = a00.z - a00.w;
    const float s01x = a01.x + a01.y, s01y = a01.z + a01.w;
    const float q01x = a01.x - a01.y, q01y = a01.z - a01.w;
    const float s10x = a10.x + a10.y, s10y = a10.z + a10.w;
    const float q10x = a10.x - a10.y, q10y = a10.z - a10.w;
    const float s11x = a11.x + a11.y, s11y = a11.z + a11.w;
    const float q11x = a11.x - a11.y, q11y = a11.z - a11.w;

    // H-axis butterfly
    const float lsx0 = s00x + s01x, lsy0 = s00y + s01y;   // lowH(sum),  d0
    const float hsx0 = s00x - s01x, hsy0 = s00y - s01y;   // highH(sum), d0
    const float lqx0 = q00x + q01x, lqy0 = q00y + q01y;   // lowH(diff), d0
    const float hqx0 = q00x - q01x, hqy0 = q00y - q01y;   // highH(diff),d0
    const float lsx1 = s10x + s11x, lsy1 = s10y + s11y;
    const float hsx1 = s10x - s11x, hsy1 = s10y - s11y;
    const float lqx1 = q10x + q11x, lqy1 = q10y + q11y;
    const float hqx1 = q10x - q11x, hqy1 = q10y - q11y;

    // D-axis butterfly + (1/sqrt2)^3 scale.  Name XYZ: X=D, Y=H, Z=W.
    const float c = 0.35355339059327373f;
    const float2 LLL = make_float2((lsx0 + lsx1) * c, (lsy0 + lsy1) * c);
    const float2 LLH = make_float2((lqx0 + lqx1) * c, (lqy0 + lqy1) * c);
    const float2 LHL = make_float2((hsx0 + hsx1) * c, (hsy0 + hsy1) * c);
    const float2 LHH = make_float2((hqx0 + hqx1) * c, (hqy0 + hqy1) * c);
    const float2 HLL = make_float2((lsx0 - lsx1) * c, (lsy0 - lsy1) * c);
    const float2 HLH = make_float2((lqx0 - lqx1) * c, (lqy0 - lqy1) * c);
    const float2 HHL = make_float2((hsx0 - hsx1) * c, (hsy0 - hsy1) * c);
    const float2 HHH = make_float2((hqx0 - hqx1) * c, (hqy0 - hqy1) * c);

    // ---- stores ----
    const size_t chanStride = (size_t)DOUT_ * HOUT_ * WOUT_;        // 65536
    const size_t inner = ((size_t)dz * HOUT_ + dyG) * WOUT_ + 2 * p;
    const int b = bc >> 5;
    const int ch = bc & 31;

    // LLL: [B,32,16,64,64] at offset 0
    *(float2*)(out + (size_t)bc * chanStride + inner) = LLL;

    // H_all: [B,224,16,64,64] at offset 64*chanStride; channel = s*32 + ch
    float* outH = out + (size_t)BC_ * chanStride
                      + (size_t)(b * 224 + ch) * chanStride + inner;
    const size_t sStride = (size_t)32 * chanStride;                 // 2097152
    *(float2*)(outH + 0 * sStride) = LLH;
    *(float2*)(outH + 1 * sStride) = LHL;
    *(float2*)(outH + 2 * sStride) = LHH;
    *(float2*)(outH + 3 * sStride) = HLL;
    *(float2*)(outH + 4 * sStride) = HLH;
    *(float2*)(outH + 5 * sStride) = HHL;
    *(float2*)(outH + 6 * sStride) = HHH;
}

extern "C" void kernel_launch(void* const* d_in, const int* in_sizes, int n_in,
                              void* d_out, int out_size, void* d_ws, size_t ws_size,
                              hipStream_t stream) {
    (void)in_sizes; (void)n_in; (void)out_size; (void)d_ws; (void)ws_size;
    const float* x = (const float*)d_in[0];
    float* out = (float*)d_out;
    // 64 (B*C) * 16 (D_out) * 8 (H tiles) = 8192 blocks of 256 threads
    haar3d_kernel<<<dim3(8192), dim3(256), 0, stream>>>(x, out);
}